// Encoder_1400159339185
// MI455X (gfx1250) — compile-verified
//
#include <hip/hip_runtime.h>

typedef __attribute__((ext_vector_type(2))) float v2f;
typedef __attribute__((ext_vector_type(8))) float v8f;

#define FEAT   128
#define S1n    10
#define S2n    10
#define NCLIPS 8
#define NSEG   3
#define OUTROW (NCLIPS * NSEG * FEAT)   // 3072 floats per node row

// ---------------------------------------------------------------------------
// Phase 1: neighbor aggregation.
// One block per flat node (128 threads = 4 waves, thread t = feature dim t).
// Indices staged in LDS so the feature gathers are the only vmem traffic.
// Writes aggF[seg][n][128] with seg: 0=feats1(mean10), 1=feats2(mean100), 2=self.
// ---------------------------------------------------------------------------
__global__ void __launch_bounds__(128) agg_kernel(
    const float* __restrict__ features,
    const int*   __restrict__ nodes,
    const int*   __restrict__ neigh1,
    const int*   __restrict__ neigh2,
    float*       __restrict__ aggF,
    int N)
{
    __shared__ int s_nf;
    __shared__ int s_idx1[S1n];
    __shared__ int s_m2[S2n];
    __shared__ int s_idx2[S1n * S2n];

    const int n = blockIdx.x;
    const int t = threadIdx.x;

    if (t == 0) s_nf = nodes[n];
    __syncthreads();
    const int nf = s_nf;

    if (t < S1n) s_idx1[t] = neigh1[nf * S1n + t];
    if (t < S2n) s_m2[t]   = neigh2[nf * S2n + t];
    __syncthreads();
    if (t < S1n * S2n)
        s_idx2[t] = neigh1[s_m2[t / S1n] * S1n + (t % S1n)];
    __syncthreads();

    // t indexes the feature dimension (blockDim.x == FEAT)
    const float selfv = features[(long)nf * FEAT + t];

    float sum1 = 0.f;
#pragma unroll
    for (int s = 0; s < S1n; ++s)
        sum1 += features[(long)s_idx1[s] * FEAT + t];

    float sum2 = 0.f;
#pragma unroll 10
    for (int s = 0; s < S1n * S2n; ++s)
        sum2 += features[(long)s_idx2[s] * FEAT + t];

    const long segStride = (long)N * FEAT;
    aggF[0 * segStride + (long)n * FEAT + t] = sum1 * (1.0f / S1n);
    aggF[1 * segStride + (long)n * FEAT + t] = sum2 * (1.0f / (S1n * S2n));
    aggF[2 * segStride + (long)n * FEAT + t] = selfv;
}

// ---------------------------------------------------------------------------
// Phase 2: per-clip projection via V_WMMA_F32_16X16X4_F32 + fused ReLU.
// Block = 256 threads (8 waves) handles 16 nodes. LDS holds the 3x16x128 fp32
// feature tile (24 KB). Each wave computes 16x16 output tiles over the
// flattened [1024 x 128] weight matrix; K loop = 32 WMMA(16x16x4) steps.
// out[n][c][seg*128 + k] = relu( sum_d W[c][k][d] * f_seg[n][d] ).
// ---------------------------------------------------------------------------
__global__ void __launch_bounds__(256) proj_kernel(
    const float* __restrict__ aggF,   // [3][N][128]
    const float* __restrict__ W,      // [1024][128] row-major (c*128+k, d)
    float*       __restrict__ out,    // [N][8][384]
    int N)
{
    __shared__ float ldsF[NSEG * 16 * FEAT];   // 24 KB

    const int  nodeBase  = blockIdx.x * 16;
    const int  tid       = threadIdx.x;
    const long segStride = (long)N * FEAT;

    // Cooperative load of the 3 x 16 x 128 fp32 tile as float4 (b128 loads).
    const int total4 = NSEG * 16 * FEAT / 4;   // 1536
    for (int i4 = tid; i4 < total4; i4 += 256) {
        const int i   = i4 * 4;
        const int seg = i / (16 * FEAT);
        const int rem = i % (16 * FEAT);
        const int row = rem / FEAT;
        const int d   = rem % FEAT;
        const float4 v = *(const float4*)&aggF[seg * segStride +
                                               (long)(nodeBase + row) * FEAT + d];
        *(float4*)&ldsF[i] = v;
    }
    __syncthreads();

    const int wave  = tid >> 5;
    const int lane  = tid & 31;
    const int laneN = lane & 15;          // A:M, B:N, D:N index
    const int laneK = (lane >> 4) * 2;    // K sub-offset per lane half

    // 192 tiles total: 3 segs x 64 ck-tiles (1024/16); 24 tiles per wave.
    for (int t = wave; t < NSEG * 64; t += 8) {
        const int seg    = t >> 6;
        const int ckBase = (t & 63) << 4;

        const float* ldsA = &ldsF[seg * 16 * FEAT + laneN * FEAT];
        const float* wRow = &W[(long)(ckBase + laneN) * FEAT];

        v8f acc = {0.f, 0.f, 0.f, 0.f, 0.f, 0.f, 0.f, 0.f};
#pragma unroll
        for (int kk = 0; kk < FEAT; kk += 4) {
            // A(16x4 f32): lanes 0-15 K={kk,kk+1}; lanes 16-31 K={kk+2,kk+3}
            v2f a = *(const v2f*)&ldsA[kk + laneK];
            // B(4x16 f32): VGPR v, lane half selects K row; N = laneN
            v2f b = *(const v2f*)&wRow[kk + laneK];
            acc = __builtin_amdgcn_wmma_f32_16x16x4_f32(
                /*neg_a=*/false, a, /*neg_b=*/false, b,
                /*c_mod=*/(short)0, acc, /*reuse_a=*/false, /*reuse_b=*/false);
        }

        // D layout: VGPR r -> M = r + 8*(lane>=16); N = laneN. Fused ReLU.
        const int ck      = ckBase + laneN;
        const int c       = ck >> 7;
        const int k       = ck & 127;
        const int colOff  = c * (NSEG * FEAT) + seg * FEAT + k;
        const int rowHalf = (lane >> 4) * 8;
#pragma unroll
        for (int r = 0; r < 8; ++r) {
            const long row = nodeBase + rowHalf + r;
            const float v  = acc[r];
            out[row * (long)OUTROW + colOff] = v > 0.f ? v : 0.f;
        }
    }
}

// ---------------------------------------------------------------------------
extern "C" void kernel_launch(void* const* d_in, const int* in_sizes, int n_in,
                              void* d_out, int out_size, void* d_ws, size_t ws_size,
                              hipStream_t stream)
{
    const float* features     = (const float*)d_in[0];  // [100000,128] f32
    const float* local_weight = (const float*)d_in[1];  // [8,128,128]  f32
    const int*   nodes        = (const int*)d_in[2];    // [1024,8]     i32
    const int*   neigh1       = (const int*)d_in[3];    // [100000,10]  i32
    const int*   neigh2       = (const int*)d_in[4];    // [100000,10]  i32

    const int N = in_sizes[2];          // B*L = 8192 flat nodes
    float* aggF = (float*)d_ws;         // [3][N][128] fp32 scratch (12.6 MB)

    agg_kernel<<<N, 128, 0, stream>>>(features, nodes, neigh1, neigh2, aggF, N);
    proj_kernel<<<N / 16, 256, 0, stream>>>(aggF, local_weight, (float*)d_out, N);
}